// CPCI_Action_30_46102178955771
// MI455X (gfx1250) — compile-verified
//
#include <hip/hip_runtime.h>

typedef float v8f __attribute__((ext_vector_type(8)));
typedef float v2f __attribute__((ext_vector_type(2)));

#define Hdim 1024
#define Zdim 1024
#define Fh   30
#define DIN  2048
#define DHh  512
#define EINd 1118
#define EHd  512
#define EOUTd 94

// ---------------- GRU step: h_new = GRUCell(x_t, h_prev) -------------------
// grid 64 blocks x 256 threads. Block b handles hidden indices [b*16, b*16+16).
// 48 row-dots (3 gates x 16 j) per block, 6 dots per wave (8 waves), lane-parallel.
__global__ __launch_bounds__(256)
void gru_step_k(int t, const float* __restrict__ hprev,
                const float* __restrict__ a_t,
                const float* __restrict__ Wih, const float* __restrict__ Whh,
                const float* __restrict__ bih, const float* __restrict__ bhh,
                float* __restrict__ a_gru)
{
    __shared__ float hL[Hdim];
    __shared__ float ghL[48];
    const int tid = threadIdx.x;
    for (int i = tid; i < Hdim; i += 256) hL[i] = hprev[i];
    __syncthreads();

    const int wave = tid >> 5, lane = tid & 31;
    const int j0 = blockIdx.x * 16;
    for (int d = wave * 6; d < wave * 6 + 6; ++d) {
        const int gate = d >> 4, jj = d & 15;
        const int row = gate * Hdim + j0 + jj;
        const float* wrow = Whh + (size_t)row * Hdim;
        float acc = 0.f;
        for (int k = lane; k < Hdim; k += 32) acc += wrow[k] * hL[k];
        for (int off = 16; off; off >>= 1) acc += __shfl_down(acc, off, 32);
        if (lane == 0) ghL[d] = acc;
    }
    __syncthreads();

    if (tid < 16) {
        const int j = j0 + tid;
        const float x0 = a_t[t * 4 + 0], x1 = a_t[t * 4 + 1];
        const float x2 = a_t[t * 4 + 2], x3 = a_t[t * 4 + 3];
        float gi[3], gh[3];
        for (int gate = 0; gate < 3; ++gate) {
            const int row = gate * Hdim + j;
            gi[gate] = bih[row] + Wih[row * 4 + 0] * x0 + Wih[row * 4 + 1] * x1
                                + Wih[row * 4 + 2] * x2 + Wih[row * 4 + 3] * x3;
            gh[gate] = ghL[gate * 16 + tid] + bhh[row];
        }
        const float r = 1.f / (1.f + __expf(-(gi[0] + gh[0])));
        const float z = 1.f / (1.f + __expf(-(gi[1] + gh[1])));
        const float n = tanhf(gi[2] + r * gh[2]);
        a_gru[t * Hdim + j] = (1.f - z) * n + z * hL[j];
    }
}

// ------------- Expert layer 1 with V_WMMA_F32_16X16X4_F32 ------------------
// grid (8, 30), 128 threads (4 waves). blockIdx.y = expert f.
// Wave computes 16 hidden cols for both pos & neg rows via WMMA:
//   D[m][n] = sum_k A[m][k]*B[k][n],  A[m][k] = W1[f, k, hbase+m],
//   B[k][0] = x_pos[k], B[k][1] = x_neg[k], other columns 0.
// LDS holds three rows {x_pos, x_neg, zeros}; each lane reads its row
// (row = min(N,2)) so no per-iteration masking VALU is needed.
__global__ __launch_bounds__(128)
void expert_l1_k(const float* __restrict__ z_pos, const float* __restrict__ z_neg,
                 const float* __restrict__ a_gru,
                 const float* __restrict__ W1, const float* __restrict__ b1,
                 float* __restrict__ h_pos, float* __restrict__ h_neg)
{
    __shared__ float xb[3 * DIN];          // [0]=pos, [1]=neg, [2]=zeros
    const int f = blockIdx.y;
    const int tid = threadIdx.x;
    for (int i = tid; i < Zdim; i += 128) {
        xb[i]              = z_pos[f * Zdim + i];
        xb[DIN + i]        = z_neg[f * Zdim + i];
        const float ag = a_gru[f * Hdim + i];
        xb[Zdim + i]       = ag;
        xb[DIN + Zdim + i] = ag;
        xb[2 * DIN + i]        = 0.f;
        xb[2 * DIN + Zdim + i] = 0.f;
    }
    __syncthreads();

    const int wave = tid >> 5, lane = tid & 31;
    const int hbase = blockIdx.x * 64 + wave * 16;
    const int n    = lane & 15;          // A row index M / D column index N
    const int kofs = (lane >> 4) * 2;    // lane half -> K sub-offset
    const float* Wf = W1 + (size_t)f * DIN * DHh;
    const float* brow = xb + (size_t)(n < 2 ? n : 2) * DIN;

    v8f c = {};
    for (int k = 0; k < DIN; k += 4) {
        v2f a, b;
        a.x = Wf[(size_t)(k + kofs)     * DHh + hbase + n];
        a.y = Wf[(size_t)(k + kofs + 1) * DHh + hbase + n];
        b.x = brow[k + kofs];
        b.y = brow[k + kofs + 1];
        c = __builtin_amdgcn_wmma_f32_16x16x4_f32(false, a, false, b,
                                                  (short)0, c, false, false);
    }

    // D: lane half h holds M = h*8+g in c[g]; N = lane&15. Keep N=0 (pos), N=1 (neg).
    const int mrow = (lane >> 4) * 8;
    if (n < 2) {
        float* dst = (n == 0) ? h_pos : h_neg;
        for (int g = 0; g < 8; ++g) {
            const int hcol = hbase + mrow + g;
            const float v = c[g] + b1[f * DHh + hcol];
            dst[f * DHh + hcol] = v > 0.f ? v : 0.f;
        }
    }
}

// ------------- Expert layer 2: logits = h . W2[f] + b2[f] ------------------
__global__ __launch_bounds__(64)
void expert_l2_k(const float* __restrict__ h_pos, const float* __restrict__ h_neg,
                 const float* __restrict__ W2, const float* __restrict__ b2,
                 float* __restrict__ out)
{
    const int f = blockIdx.x;
    const int wave = threadIdx.x >> 5, lane = threadIdx.x & 31;
    const float* hb = wave ? h_neg : h_pos;
    float acc = 0.f;
    for (int hh = lane; hh < DHh; hh += 32)
        acc += hb[f * DHh + hh] * W2[f * DHh + hh];
    for (int off = 16; off; off >>= 1) acc += __shfl_down(acc, off, 32);
    if (lane == 0) out[wave * Fh + f] = acc + b2[f];
}

// --------------------------- evalMLP ---------------------------------------
__global__ __launch_bounds__(32)
void eval_l1_k(const float* __restrict__ b_t, const float* __restrict__ ipo,
               const float* __restrict__ W1, const float* __restrict__ b1,
               float* __restrict__ eh)
{
    const int col = blockIdx.x * 32 + threadIdx.x;
    float acc = b1[col];
    for (int i = 0; i < Hdim; ++i)  acc += b_t[i] * W1[(size_t)i * EHd + col];
    for (int i = 0; i < EOUTd; ++i) acc += ipo[i] * W1[(size_t)(Hdim + i) * EHd + col];
    eh[col] = acc > 0.f ? acc : 0.f;
}

__global__ __launch_bounds__(128)
void eval_l2_k(const float* __restrict__ eh, const float* __restrict__ W2,
               const float* __restrict__ b2, float* __restrict__ out)
{
    __shared__ float ehL[EHd];
    for (int i = threadIdx.x; i < EHd; i += 128) ehL[i] = eh[i];
    __syncthreads();
    const int t = threadIdx.x;
    if (t < EOUTd) {
        float acc = b2[t];
        for (int hh = 0; hh < EHd; ++hh) acc += ehL[hh] * W2[hh * EOUTd + t];
        out[2 * Fh + t] = 1.f / (1.f + __expf(-acc));
    }
}

// ---------------------------------------------------------------------------
extern "C" void kernel_launch(void* const* d_in, const int* in_sizes, int n_in,
                              void* d_out, int out_size, void* d_ws, size_t ws_size,
                              hipStream_t stream)
{
    (void)in_sizes; (void)n_in; (void)out_size; (void)ws_size;
    const float* b_t    = (const float*)d_in[0];
    const float* a_t    = (const float*)d_in[1];
    const float* z_pos  = (const float*)d_in[2];
    const float* z_neg  = (const float*)d_in[3];
    const float* ipo    = (const float*)d_in[4];
    const float* gWih   = (const float*)d_in[5];
    const float* gWhh   = (const float*)d_in[6];
    const float* gbih   = (const float*)d_in[7];
    const float* gbhh   = (const float*)d_in[8];
    const float* mW1    = (const float*)d_in[9];
    const float* mb1    = (const float*)d_in[10];
    const float* mW2    = (const float*)d_in[11];
    const float* mb2    = (const float*)d_in[12];
    const float* eW1    = (const float*)d_in[13];
    const float* eb1    = (const float*)d_in[14];
    const float* eW2    = (const float*)d_in[15];
    const float* eb2    = (const float*)d_in[16];
    float* out = (float*)d_out;

    float* ws    = (float*)d_ws;
    float* a_gru = ws;                       // 30*1024
    float* h_pos = a_gru + Fh * Hdim;        // 30*512
    float* h_neg = h_pos + Fh * DHh;         // 30*512
    float* eh    = h_neg + Fh * DHh;         // 512

    // Sequential GRU chain (graph-capture safe: fixed 30 launches on stream).
    for (int t = 0; t < Fh; ++t) {
        const float* hprev = (t == 0) ? b_t : (a_gru + (size_t)(t - 1) * Hdim);
        gru_step_k<<<dim3(64), dim3(256), 0, stream>>>(
            t, hprev, a_t, gWih, gWhh, gbih, gbhh, a_gru);
    }

    // Experts: HBM-bandwidth-bound grouped GEMMs via fp32 WMMA.
    expert_l1_k<<<dim3(8, Fh), dim3(128), 0, stream>>>(
        z_pos, z_neg, a_gru, mW1, mb1, h_pos, h_neg);
    expert_l2_k<<<dim3(Fh), dim3(64), 0, stream>>>(h_pos, h_neg, mW2, mb2, out);

    // evalMLP (independent of GRU; serialized on stream, tiny).
    eval_l1_k<<<dim3(16), dim3(32), 0, stream>>>(b_t, ipo, eW1, eb1, eh);
    eval_l2_k<<<dim3(1), dim3(128), 0, stream>>>(eh, eW2, eb2, out);
}